// MID_LOSS_24498493457014
// MI455X (gfx1250) — compile-verified
//
#include <hip/hip_runtime.h>
#include <math.h>

#define EMBED 512
#define NFEA  8
#define NCLS  14
#define BATCH 8192
#define BETA  0.3f

typedef __attribute__((ext_vector_type(2))) float v2f;
typedef __attribute__((ext_vector_type(8))) float v8f;

__device__ __forceinline__ float softplusf(float v) {
    // numerically stable softplus: max(v,0) + log1p(exp(-|v|))
    return fmaxf(v, 0.f) + log1pf(__expf(-fabsf(v)));
}

// Kernel 1: Gram matrix G[c][c'] = sum_e w[e,c]*w[e,c'] (padded 16x16), and zero d_out.
__global__ void gram_kernel(const float* __restrict__ w, float* __restrict__ G,
                            float* __restrict__ out) {
    int t = threadIdx.x;
    if (t == 0) out[0] = 0.f;
    if (t < NCLS * NCLS) {
        int c = t / NCLS, c2 = t % NCLS;
        float s = 0.f;
        for (int e = 0; e < EMBED; ++e)
            s = fmaf(w[e * NCLS + c], w[e * NCLS + c2], s);
        G[c * 16 + c2] = s;
    }
}

// Kernel 2: main loss. 256 threads = 8 waves; each wave owns 2 batches (16x16 WMMA tile).
__global__ void __launch_bounds__(256, 2) loss_kernel(
    const float* __restrict__ x, const int* __restrict__ y,
    const float* __restrict__ w, const float* __restrict__ G,
    float* __restrict__ out)
{
    // B fragments of w in exact WMMA layout: [step s][lane][2 floats], C padded 14->16 with zeros.
    __shared__ float wfrag[128 * 64];  // 32 KB

    const int tid = threadIdx.x;
    for (int idx = tid; idx < 128 * 64; idx += 256) {
        int s  = idx >> 6;
        int ln = (idx >> 1) & 31;
        int j  = idx & 1;
        int c  = ln & 15;
        int oo = (ln >> 4) << 1;           // lanes 16-31 hold K-rows +2
        int e  = 4 * s + oo + j;
        wfrag[idx] = (c < NCLS) ? w[e * NCLS + c] : 0.f;
    }
    __syncthreads();

    const int wave = tid >> 5;
    const int lane = tid & 31;
    const int b0   = blockIdx.x * 16 + wave * 2;

    const int r    = lane & 15;   // A-matrix row within tile
    const int half = lane >> 4;   // 0: K rows 0,1 / batch b0 cols ; 1: K rows 2,3
    const int m    = r & 7;       // feature index
    const int bsel = r >> 3;      // which of the 2 batches this A-row belongs to
    const int o    = half << 1;

    // A[row=r, e] = x[b, e*8 + m]; fold o into base; stride per K-step (4 e's) = 32 floats
    const float* xa = x + (size_t)(b0 + bsel) * (EMBED * NFEA) + (size_t)o * NFEA + m;
    __builtin_prefetch(xa, 0, 3);

    v8f acc = {};
    const v2f* bptr = (const v2f*)wfrag + lane;   // wfrag as [128][32] x v2f

    #pragma unroll 4
    for (int s = 0; s < 128; ++s) {
        float a0 = xa[0];
        float a1 = xa[NFEA];
        v2f a = {a0, a1};
        v2f bv = bptr[s * 32];
        // D = A(16x4 f32) * B(4x16 f32) + C
        acc = __builtin_amdgcn_wmma_f32_16x16x4_f32(
            false, a, false, bv, (short)0, acc, false, false);
        xa += 32;
    }

    // dot[b, c] = max over m: registers g=0..7 are m=0..7 of this lane's batch/column.
    float dotv = acc[0];
    dotv = fmaxf(dotv, acc[1]); dotv = fmaxf(dotv, acc[2]);
    dotv = fmaxf(dotv, acc[3]); dotv = fmaxf(dotv, acc[4]);
    dotv = fmaxf(dotv, acc[5]); dotv = fmaxf(dotv, acc[6]);
    dotv = fmaxf(dotv, acc[7]);

    const int c = r;               // class index for this lane
    const int b = b0 + half;
    int yv = (c < NCLS) ? y[b * NCLS + c] : 0;
    unsigned long long bal = __ballot(yv == 1);
    unsigned pmask = ((unsigned)(bal >> (half * 16))) & 0x3FFFu;
    unsigned nmask = (~pmask) & 0x3FFFu;
    int npos = __popc(pmask);
    int nneg = __popc(nmask);
    bool isP = (c < NCLS) && ((pmask >> c) & 1u);

    // pairwise softplus: lane = positive class p=c; gather dot[b,n] via shuffles (convergent)
    float spair = 0.f;
    #pragma unroll
    for (int n = 0; n < NCLS; ++n) {
        float dn = __shfl(dotv, (lane & 16) + n, 32);
        if (isP && ((nmask >> n) & 1u)) spair += softplusf(dn - dotv);
    }
    float snone = isP ? softplusf(-dotv) : 0.f;

    // diversity via Gram matrix
    float grow = 0.f, gdiag = 0.f;
    if (isP) {
        gdiag = G[c * 16 + c];
        #pragma unroll
        for (int c2 = 0; c2 < NCLS; ++c2)
            if ((pmask >> c2) & 1u) grow += G[c * 16 + c2];
    }

    // reduce within each 16-lane half (xor masks < 16 stay in-half)
    #pragma unroll
    for (int off = 8; off >= 1; off >>= 1) {
        spair += __shfl_xor(spair, off, 32);
        snone += __shfl_xor(snone, off, 32);
        grow  += __shfl_xor(grow,  off, 32);
        gdiag += __shfl_xor(gdiag, off, 32);
    }

    float S      = (nneg > 0) ? spair : snone;
    float base_l = S / (float)npos;
    float tv     = (npos > 1) ? (gdiag - grow / (float)npos) / (float)(npos - 1) : 0.f;

    // l1_err: coalesced second pass (hits L2), lane-local variance over the 8 contiguous m's
    float l1v[2];
    #pragma unroll
    for (int bi = 0; bi < 2; ++bi) {
        const float* xb = x + (size_t)(b0 + bi) * (EMBED * NFEA);
        float a1s = 0.f;
        for (int e = lane; e < EMBED; e += 32) {
            const float4* p = (const float4*)(xb + (size_t)e * NFEA);
            float4 u0 = p[0], u1 = p[1];
            float s1 = (u0.x + u0.y) + (u0.z + u0.w) + (u1.x + u1.y) + (u1.z + u1.w);
            float s2 = fmaf(u0.x, u0.x, fmaf(u0.y, u0.y, fmaf(u0.z, u0.z, u0.w * u0.w)))
                     + fmaf(u1.x, u1.x, fmaf(u1.y, u1.y, fmaf(u1.z, u1.z, u1.w * u1.w)));
            a1s += fabsf((s2 - s1 * s1 * 0.125f) * (1.f / 7.f));
        }
        #pragma unroll
        for (int off = 16; off >= 1; off >>= 1)
            a1s += __shfl_xor(a1s, off, 32);
        l1v[bi] = a1s;
    }

    if ((lane & 15) == 0) {  // lane 0 -> batch b0, lane 16 -> batch b0+1
        float loss_b = 2.f * ((1.f - BETA) * (1.f + tv) * base_l + BETA * l1v[half]);
        atomicAdd(out, loss_b * (1.f / (float)BATCH));
    }
}

extern "C" void kernel_launch(void* const* d_in, const int* in_sizes, int n_in,
                              void* d_out, int out_size, void* d_ws, size_t ws_size,
                              hipStream_t stream) {
    const float* x = (const float*)d_in[0];
    const int*   y = (const int*)d_in[1];
    const float* w = (const float*)d_in[2];
    float* out = (float*)d_out;
    float* G   = (float*)d_ws;           // 256 floats

    gram_kernel<<<1, 256, 0, stream>>>(w, G, out);
    loss_kernel<<<BATCH / 16, 256, 0, stream>>>(x, y, w, G, out);
}